// GCNConv_68049461837963
// MI455X (gfx1250) — compile-verified
//
#include <hip/hip_runtime.h>
#include <hip/hip_bf16.h>

// GCNConv on MI455X (gfx1250, wave32):
//   out = D^-1/2 (A+I) D^-1/2 X W + b
// Phase 1: degree + rsqrt       (int atomics)
// Phase 2: h = X @ W            (V_WMMA_F32_16X16X4_F32, exact fp32)
// Phase 3: out = h*dinv^2 + b   (self loops, also fully inits d_out)
// Phase 4: edge scatter         (global_atomic_add_f32, one wave per edge)

typedef float v2f __attribute__((ext_vector_type(2)));
typedef float v8f __attribute__((ext_vector_type(8)));

#define NF   128        // NIN == NOUT == 128
#define NF4  (NF / 4)   // float4 chunks per row (32 == wave width)

// ---------------------------------------------------------------- degrees ---
__global__ __launch_bounds__(256) void deg_init_kernel(int* __restrict__ deg, int n) {
  int i = blockIdx.x * blockDim.x + threadIdx.x;
  if (i < n) deg[i] = 1;                       // self-loop
}

__global__ __launch_bounds__(256) void deg_count_kernel(const int* __restrict__ dst,
                                                        int e, int* __restrict__ deg) {
  int stride = gridDim.x * blockDim.x;
  for (int i = blockIdx.x * blockDim.x + threadIdx.x; i < e; i += stride)
    atomicAdd(&deg[dst[i]], 1);
}

__global__ __launch_bounds__(256) void dinv_kernel(const int* __restrict__ deg,
                                                   float* __restrict__ dinv, int n) {
  int i = blockIdx.x * blockDim.x + threadIdx.x;
  if (i < n) dinv[i] = rsqrtf((float)deg[i]);  // deg >= 1 always (self loop)
}

// ------------------------------------------------------------- WMMA GEMM ----
// Block = 256 threads = 8 waves. Wave w owns output columns [16w, 16w+16).
// B fragments (W) preloaded to registers once; X tiles streamed via LDS.
// A-frag (16x4 f32): lane<16 -> {K=0,K=1} of row M=lane; lane>=16 -> {K=2,K=3}.
// B-frag (4x16 f32): lane<16 -> {K=0,K=1} of col N=lane; lane>=16 -> {K=2,K=3}.
// C/D (16x16 f32):   VGPR v = row (v + 8*(lane>=16)), col = lane&15.
__global__ __launch_bounds__(256) void gemm_wmma_kernel(const float* __restrict__ x,
                                                        const float* __restrict__ W,
                                                        float* __restrict__ h,
                                                        int nTiles) {
  __shared__ float xs[16][NF + 4];             // +4 pad: conflict-free col reads

  const int lane = threadIdx.x & 31;
  const int wave = threadIdx.x >> 5;           // 0..7
  const int c0   = wave * 16;
  const int half = lane >> 4;                  // 0: K pair {0,1}, 1: {2,3}
  const int lm   = lane & 15;

  // Preload all 32 B fragments of W for this wave's 16-column stripe.
  v2f bfrag[32];
#pragma unroll
  for (int s = 0; s < 32; ++s) {
    const int k = 4 * s + 2 * half;
    v2f bf;
    bf.x = W[(k + 0) * NF + c0 + lm];
    bf.y = W[(k + 1) * NF + c0 + lm];
    bfrag[s] = bf;
  }

  // Cooperative x-tile load mapping: 2048 floats / 256 threads = 8 each.
  const int flat = threadIdx.x * 8;
  const int lrow = flat >> 7;                  // /128
  const int lcol = flat & (NF - 1);

  for (int tile = blockIdx.x; tile < nTiles; tile += gridDim.x) {
    const int r0 = tile * 16;
    const float4* g4 = (const float4*)(x + (size_t)(r0 + lrow) * NF + lcol);
    const float4 a0 = g4[0];
    const float4 a1 = g4[1];

    __syncthreads();                           // prior readers done with xs
    *(float4*)&xs[lrow][lcol]     = a0;
    *(float4*)&xs[lrow][lcol + 4] = a1;
    __syncthreads();

    v8f acc = {};
#pragma unroll
    for (int s = 0; s < 32; ++s) {
      const int k = 4 * s + 2 * half;
      v2f af;
      af.x = xs[lm][k];
      af.y = xs[lm][k + 1];
      acc = __builtin_amdgcn_wmma_f32_16x16x4_f32(
          /*neg_a=*/false, af, /*neg_b=*/false, bfrag[s],
          /*c_mod=*/(short)0, acc, /*reuse_a=*/false, /*reuse_b=*/false);
    }

    float* op = h + (size_t)r0 * NF + c0 + lm;
#pragma unroll
    for (int v = 0; v < 8; ++v)
      op[(size_t)(half * 8 + v) * NF] = acc[v];
  }
}

// Scalar fallback for a row-count tail (unused when N % 16 == 0).
__global__ __launch_bounds__(128) void gemm_tail_kernel(const float* __restrict__ x,
                                                        const float* __restrict__ W,
                                                        float* __restrict__ h,
                                                        int rowStart, int nRows) {
  const int row = rowStart + blockIdx.x;
  if (row >= nRows) return;
  const int c = threadIdx.x;
  float acc = 0.f;
  for (int k = 0; k < NF; ++k)
    acc = fmaf(x[(size_t)row * NF + k], W[k * NF + c], acc);
  h[(size_t)row * NF + c] = acc;
}

// ------------------------------------------------- self-loop + bias init ----
__global__ __launch_bounds__(256) void out_init_kernel(const float* __restrict__ h,
                                                       const float* __restrict__ dinv,
                                                       const float* __restrict__ bias,
                                                       float* __restrict__ out, int n) {
  const int idx = blockIdx.x * blockDim.x + threadIdx.x;   // n * 32 items
  if (idx >= n * NF4) return;
  const int node = idx >> 5;                               // / NF4
  const int q    = idx & (NF4 - 1);
  float s = dinv[node];
  s *= s;                                                  // dinv[i]^2 (self loop)
  const float4 hv = ((const float4*)h)[(size_t)node * NF4 + q];
  const float4 bv = ((const float4*)bias)[q];
  float4 o;
  o.x = fmaf(hv.x, s, bv.x);
  o.y = fmaf(hv.y, s, bv.y);
  o.z = fmaf(hv.z, s, bv.z);
  o.w = fmaf(hv.w, s, bv.w);
  ((float4*)out)[(size_t)node * NF4 + q] = o;
}

// ----------------------------------------------------------- edge scatter ---
// One wave per edge: lane l handles float4 chunk l of the 128-wide feature row.
__global__ __launch_bounds__(256) void scatter_kernel(const int* __restrict__ src,
                                                      const int* __restrict__ dst,
                                                      const float* __restrict__ dinv,
                                                      const float* __restrict__ h,
                                                      float* __restrict__ out, int E) {
  const int gtid = blockIdx.x * blockDim.x + threadIdx.x;
  const int lane = gtid & 31;
  const int wid  = gtid >> 5;
  const int nw   = (gridDim.x * blockDim.x) >> 5;
  for (int e = wid; e < E; e += nw) {
    if (e + nw < E) {                          // gfx1250 global_prefetch_b8
      __builtin_prefetch(&src[e + nw], 0, 0);
      __builtin_prefetch(&dst[e + nw], 0, 0);
    }
    const int s = src[e];
    const int d = dst[e];
    const float norm = dinv[s] * dinv[d];
    const float4 hv = ((const float4*)h)[(size_t)s * NF4 + lane];
    float* op = out + (size_t)d * NF + lane * 4;
    unsafeAtomicAdd(op + 0, hv.x * norm);      // -> global_atomic_add_f32
    unsafeAtomicAdd(op + 1, hv.y * norm);
    unsafeAtomicAdd(op + 2, hv.z * norm);
    unsafeAtomicAdd(op + 3, hv.w * norm);
  }
}

// ------------------------------------------------------------------ launch --
extern "C" void kernel_launch(void* const* d_in, const int* in_sizes, int n_in,
                              void* d_out, int out_size, void* d_ws, size_t ws_size,
                              hipStream_t stream) {
  (void)n_in; (void)out_size; (void)ws_size;

  const float* x          = (const float*)d_in[0];   // [N, 128]
  const int*   edge_index = (const int*)  d_in[1];   // [2, E]
  /* d_in[2] = edge_attr: accepted but ignored by reference */
  const float* W          = (const float*)d_in[3];   // [128, 128]
  const float* bias       = (const float*)d_in[4];   // [128]
  float*       out        = (float*)d_out;           // [N, 128]

  const int N = in_sizes[0] / NF;
  const int E = in_sizes[1] / 2;
  const int* src = edge_index;
  const int* dst = edge_index + E;

  // Workspace carve-out (256B aligned): deg[N] int, dinv[N] f32, h[N*128] f32.
  char* base = (char*)d_ws;
  size_t off = 0;
  int* deg = (int*)(base + off);
  off = (off + (size_t)N * sizeof(int)   + 255) & ~(size_t)255;
  float* dinv = (float*)(base + off);
  off = (off + (size_t)N * sizeof(float) + 255) & ~(size_t)255;
  float* h = (float*)(base + off);

  deg_init_kernel<<<(N + 255) / 256, 256, 0, stream>>>(deg, N);
  deg_count_kernel<<<2048, 256, 0, stream>>>(dst, E, deg);
  dinv_kernel<<<(N + 255) / 256, 256, 0, stream>>>(deg, dinv, N);

  const int nTiles = N / 16;
  if (nTiles > 0) {
    const int blocks = nTiles < 1280 ? nTiles : 1280;
    gemm_wmma_kernel<<<blocks, 256, 0, stream>>>(x, W, h, nTiles);
  }
  const int rem = N % 16;
  if (rem > 0)
    gemm_tail_kernel<<<rem, NF, 0, stream>>>(x, W, h, nTiles * 16, N);

  out_init_kernel<<<(N * NF4 + 255) / 256, 256, 0, stream>>>(h, dinv, bias, out, N);
  scatter_kernel<<<4096, 256, 0, stream>>>(src, dst, dinv, h, out, E);
}